// PsiU_23493471109190
// MI455X (gfx1250) — compile-verified
//
#include <hip/hip_runtime.h>
#include <math.h>

// Problem constants (from the reference)
#define Nc   128   // N
#define Mc   128   // M
#define NXIc 256
#define Lc   256
#define HDc  768   // 2*NXI + L
#define EPS_REG 0.001f

typedef __attribute__((ext_vector_type(2))) float v2f;
typedef __attribute__((ext_vector_type(8))) float v8f;

// ---------------------------------------------------------------------------
// Small generic fp32 GEMM for the derived-parameter phase (matrices <= 768).
// C[i,j] = alpha * sum_k opA(i,k)*opB(k,j) + beta*C[i,j] (+ diagonal add)
// alphaMode: 0 -> aConst ; 1 -> aConst*gamma ; 2 -> aConst/gamma  (gamma=sg^2)
// diagMode : 0 none ; 1 add EPS_REG on diagonal ; 2 add gamma on diagonal
// ---------------------------------------------------------------------------
__global__ void gemm_s(float* C, int ldc, const float* A, int lda,
                       const float* B, int ldb,
                       int M, int N, int K, int tA, int tB,
                       float aConst, int aMode, const float* sgp,
                       float beta, int diagMode)
{
    int idx = blockIdx.x * blockDim.x + threadIdx.x;
    if (idx >= M * N) return;
    int i = idx / N, j = idx - i * N;
    float s = 0.0f;
    for (int k = 0; k < K; ++k) {
        float a = tA ? A[(size_t)k * lda + i] : A[(size_t)i * lda + k];
        float b = tB ? B[(size_t)j * ldb + k] : B[(size_t)k * ldb + j];
        s = fmaf(a, b, s);
    }
    float gamma = sgp[0] * sgp[0];
    float alpha = aConst;
    if (aMode == 1) alpha *= gamma;
    else if (aMode == 2) alpha /= gamma;
    float r = alpha * s;
    if (beta != 0.0f) r += beta * C[(size_t)i * ldc + j];
    if (diagMode != 0 && i == j) r += (diagMode == 1) ? EPS_REG : gamma;
    C[(size_t)i * ldc + j] = r;
}

// ----------------------------- elementwise helpers -------------------------
__global__ void k_skew(float* C, const float* Ym, int n) {   // C += Y - Y^T
    int idx = blockIdx.x * blockDim.x + threadIdx.x;
    if (idx >= n * n) return;
    int i = idx / n, j = idx - i * n;
    C[idx] += Ym[idx] - Ym[j * n + i];
}
__global__ void k_A0(float* A0, const float* Mm, int n) {    // A0 = I - Mm
    int idx = blockIdx.x * blockDim.x + threadIdx.x;
    if (idx >= n * n) return;
    int i = idx / n, j = idx - i * n;
    A0[idx] = ((i == j) ? 1.0f : 0.0f) - Mm[idx];
}
__global__ void k_addT(float* C, int ldc, const float* S, int lds,
                       int M, int N, float coef) {           // C += coef * S^T
    int idx = blockIdx.x * blockDim.x + threadIdx.x;
    if (idx >= M * N) return;
    int i = idx / N, j = idx - i * N;
    C[(size_t)i * ldc + j] += coef * S[(size_t)j * lds + i];
}
__global__ void k_copyT(float* dst, const float* src, int sm, int sn) {
    // dst (sn x sm) = src(sm x sn)^T
    int idx = blockIdx.x * blockDim.x + threadIdx.x;
    if (idx >= sm * sn) return;
    int i = idx / sn, j = idx - i * sn;
    dst[(size_t)j * sm + i] = src[idx];
}
__global__ void k_copy(float* dst, const float* src, int tot) {
    int idx = blockIdx.x * blockDim.x + threadIdx.x;
    if (idx < tot) dst[idx] = src[idx];
}
__global__ void k_zero(float* dst, int tot) {
    int idx = blockIdx.x * blockDim.x + threadIdx.x;
    if (idx < tot) dst[idx] = 0.0f;
}
__global__ void k_E(float* E, const float* H, const float* Ym) {
    // E = 0.5*(H11 + H33 + Y - Y^T), 256x256
    int idx = blockIdx.x * blockDim.x + threadIdx.x;
    if (idx >= 256 * 256) return;
    int i = idx / 256, j = idx - i * 256;
    E[idx] = 0.5f * (H[(size_t)i * HDc + j] +
                     H[(size_t)(512 + i) * HDc + 512 + j] +
                     Ym[idx] - Ym[j * 256 + i]);
}

// ---------------------------------------------------------------------------
// Single-workgroup Gauss-Jordan inversion (no pivoting; matrices are
// diagonally dominant here). aug is n x 2n scratch in global memory.
// addIdent: invert (I + in) instead of (in).
// ---------------------------------------------------------------------------
__global__ void __launch_bounds__(1024)
invert_gj(float* out, const float* in, int ldin, int n, int addIdent, float* aug)
{
    __shared__ float fac[256];
    int tid = threadIdx.x, nt = blockDim.x;
    int n2 = 2 * n, tot = n * n2;
    for (int idx = tid; idx < tot; idx += nt) {
        int i = idx / n2, j = idx - i * n2;
        float v;
        if (j < n) v = in[(size_t)i * ldin + j] + ((addIdent && i == j) ? 1.0f : 0.0f);
        else       v = (j - n == i) ? 1.0f : 0.0f;
        aug[idx] = v;
    }
    __threadfence(); __syncthreads();
    for (int k = 0; k < n; ++k) {
        float pinv = 1.0f / aug[(size_t)k * n2 + k];
        __syncthreads();
        for (int j = tid; j < n2; j += nt) aug[(size_t)k * n2 + j] *= pinv;
        __threadfence(); __syncthreads();
        for (int i = tid; i < n; i += nt) fac[i] = aug[(size_t)i * n2 + k];
        __threadfence(); __syncthreads();
        for (int idx = tid; idx < tot; idx += nt) {
            int i = idx / n2, j = idx - i * n2;
            if (i != k) aug[idx] -= fac[i] * aug[(size_t)k * n2 + j];
        }
        __threadfence(); __syncthreads();
    }
    for (int idx = tid; idx < n * n; idx += nt) {
        int i = idx / n, j = idx - i * n;
        out[idx] = aug[(size_t)i * n2 + n + j];
    }
}

// ---------------------------------------------------------------------------
// Sequential tanh / triangular recurrence over L=256 (parallel over batch).
// One wave (32 threads) per block, 1 thread = 1 batch row.
//   eps[b,i] = tanh( (cx[b,i] - sum_{j<i} H22[i,j]*eps[b,j]) / H22[i,i] )
// (D11 = -tril(H22,-1), Lam = diag(H22))
// ---------------------------------------------------------------------------
__global__ void __launch_bounds__(32)
recur_eps(float* epsb, const float* cx, const float* H22)
{
    __shared__ float le[Lc * 32];   // eps tile, [i][lane]
    __shared__ float row[Lc];       // current H22 row
    int tid = threadIdx.x;
    size_t b = (size_t)blockIdx.x * 32 + tid;
    const float* cxr = cx + b * Lc;
    float* er = epsb + b * Lc;
    for (int i = 0; i < Lc; ++i) {
        for (int j = tid; j < Lc; j += 32) row[j] = H22[(size_t)i * HDc + j];
        __syncthreads();
        float acc = cxr[i];
        for (int j = 0; j < i; ++j)
            acc = fmaf(-row[j], le[j * 32 + tid], acc);
        float v = tanhf(acc / row[i]);
        le[i * 32 + tid] = v;
        er[i] = v;
        __syncthreads();
    }
}

// ---------------------------------------------------------------------------
// Batched WMMA f32 GEMM: out[b,n] = sum_s al_s * A_s[b,:] . W_s[n,:]
// (i.e. out = sum_s al_s * A_s @ W_s^T). A_s has row stride = K_s.
//
// Register-blocked: each wave computes a 32x32 macro-tile = 2x2 WMMA tiles,
// so each 4-K step issues 4 global_load_b64 and 4 v_wmma (1 load / wmma),
// with the alpha scale amortized over 2 WMMAs per B-frag.
//
// V_WMMA_F32_16X16X4_F32 fragment layout (ISA 7.12.2):
//  A-frag (16x4): lane l -> A[m=l%16][k + 2*(l>>4) + {0,1}]
//  B-frag (4x16): lane l -> W[n=l%16][k + 2*(l>>4) + {0,1}]   (mirrored)
//  C/D: VGPR r -> row m0 + r + 8*(l>>4), col n0 + l%16.
// ---------------------------------------------------------------------------
__device__ __forceinline__ void bg_term2(v8f acc[2][2], const float* A, const float* W,
                                         int ldw, int K, float al,
                                         int m0, int n0, int half, int ml)
{
    const float* ap0 = A + (size_t)(m0 + ml) * (size_t)K   + 2 * half;
    const float* ap1 = ap0 + (size_t)16 * (size_t)K;
    const float* wp0 = W + (size_t)(n0 + ml) * (size_t)ldw + 2 * half;
    const float* wp1 = wp0 + (size_t)16 * (size_t)ldw;
    for (int k = 0; k < K; k += 4) {
        v2f a0 = *(const v2f*)(ap0 + k);
        v2f a1 = *(const v2f*)(ap1 + k);
        v2f b0 = *(const v2f*)(wp0 + k);
        v2f b1 = *(const v2f*)(wp1 + k);
        b0.x *= al; b0.y *= al;
        b1.x *= al; b1.y *= al;
        acc[0][0] = __builtin_amdgcn_wmma_f32_16x16x4_f32(
            false, a0, false, b0, (short)0, acc[0][0], false, false);
        acc[0][1] = __builtin_amdgcn_wmma_f32_16x16x4_f32(
            false, a0, false, b1, (short)0, acc[0][1], false, false);
        acc[1][0] = __builtin_amdgcn_wmma_f32_16x16x4_f32(
            false, a1, false, b0, (short)0, acc[1][0], false, false);
        acc[1][1] = __builtin_amdgcn_wmma_f32_16x16x4_f32(
            false, a1, false, b1, (short)0, acc[1][1], false, false);
    }
}

__global__ void __launch_bounds__(256)
batch_gemm3(float* out, int ldo,
            const float* A0p, const float* W0, int ldw0, int K0, float al0,
            const float* A1p, const float* W1, int ldw1, int K1, float al1,
            const float* A2p, const float* W2, int ldw2, int K2, float al2)
{
    int wave = threadIdx.x >> 5;
    int lane = threadIdx.x & 31;
    // 8 waves arranged 4 (rows) x 2 (cols); block covers 128 rows x 64 cols
    int m0 = blockIdx.y * 128 + (wave >> 1) * 32;
    int n0 = blockIdx.x * 64  + (wave & 1) * 32;
    int half = lane >> 4;
    int ml   = lane & 15;
    v8f z = {0.f, 0.f, 0.f, 0.f, 0.f, 0.f, 0.f, 0.f};
    v8f acc[2][2];
    acc[0][0] = z; acc[0][1] = z; acc[1][0] = z; acc[1][1] = z;
    bg_term2(acc, A0p, W0, ldw0, K0, al0, m0, n0, half, ml);
    if (A1p) bg_term2(acc, A1p, W1, ldw1, K1, al1, m0, n0, half, ml);
    if (A2p) bg_term2(acc, A2p, W2, ldw2, K2, al2, m0, n0, half, ml);
#pragma unroll
    for (int tm = 0; tm < 2; ++tm) {
#pragma unroll
        for (int tn = 0; tn < 2; ++tn) {
            size_t rowBase = (size_t)(m0 + tm * 16 + 8 * half);
            int col = n0 + tn * 16 + ml;
#pragma unroll
            for (int r = 0; r < 8; ++r)
                out[(rowBase + r) * (size_t)ldo + col] = acc[tm][tn][r];
        }
    }
}

// ---------------------------------------------------------------------------
extern "C" void kernel_launch(void* const* d_in, const int* in_sizes, int n_in,
                              void* d_out, int out_size, void* d_ws, size_t ws_size,
                              hipStream_t stream)
{
    const float* w   = (const float*)d_in[0];   // (B, 128)
    const float* xi  = (const float*)d_in[1];   // (B, 256)
    const float* X   = (const float*)d_in[2];   // (768, 768)
    const float* Y   = (const float*)d_in[3];   // (256, 256)
    const float* B2  = (const float*)d_in[4];   // (256, 128)
    const float* C2  = (const float*)d_in[5];   // (128, 256)
    const float* D21 = (const float*)d_in[6];   // (128, 256)
    const float* X3  = (const float*)d_in[7];   // (128, 128)
    const float* Y3  = (const float*)d_in[8];   // (128, 128)
    const float* sg  = (const float*)d_in[9];   // (1,1)
    const float* D12 = (const float*)d_in[10];  // (256, 128)
    int Bb = in_sizes[0] / Nc;                  // 32768

    float* ws    = (float*)d_ws;
    float* MMp   = ws;                 // 128x128
    float* A0p   = ws + 16384;         // 128x128
    float* PINV  = ws + 32768;         // 128x128
    float* D22p  = ws + 49152;         // 128x128
    float* RCAP  = ws + 65536;         // 128x128
    float* RINV  = ws + 81920;         // 128x128
    float* C2CAP = ws + 98304;         // 128x256
    float* D21CAP= ws + 131072;        // 128x256
    float* VECR  = ws + 163840;        // 768x128
    float* VECQ  = ws + 262144;        // 768x128
    float* T1    = ws + 360448;        // 768x128
    float* Hm    = ws + 458752;        // 768x768
    float* Ep    = ws + 1048576;       // 256x256
    float* EINV  = ws + 1114112;       // 256x256
    float* AUG   = ws + 1179648;       // 256x512 GJ scratch
    float* CX    = ws + 1310720;       // B x 256 (reused as E_xi)
    float* EPSb  = CX + (size_t)Bb * Lc; // B x 256

    float* u_out  = (float*)d_out;                 // (B, 128)
    float* xn_out = u_out + (size_t)Bb * Mc;       // (B, 256)

    auto g1 = [](int tot) { return dim3((unsigned)((tot + 255) / 256)); };

    // ---- derived parameters ----
    // Mm = X3 @ X3^T + EPS*I ; then += Y3 - Y3^T
    gemm_s<<<g1(128*128),256,0,stream>>>(MMp,128, X3,128, X3,128,
                                         128,128,128, 0,1, 1.f,0,sg, 0.f, 1);
    k_skew<<<g1(128*128),256,0,stream>>>(MMp, Y3, 128);
    k_A0  <<<g1(128*128),256,0,stream>>>(A0p, MMp, 128);
    invert_gj<<<1,1024,0,stream>>>(PINV, MMp, 128, 128, 1, AUG);   // inv(I+Mm)
    // D22 = gamma * (I - Mm) @ inv(I+Mm)
    gemm_s<<<g1(128*128),256,0,stream>>>(D22p,128, A0p,128, PINV,128,
                                         128,128,128, 0,0, 1.f,1,sg, 0.f, 0);
    // R_cap = gamma*I - (1/gamma) D22^T D22 ; then invert
    gemm_s<<<g1(128*128),256,0,stream>>>(RCAP,128, D22p,128, D22p,128,
                                         128,128,128, 1,0, -1.f,2,sg, 0.f, 2);
    invert_gj<<<1,1024,0,stream>>>(RINV, RCAP, 128, 128, 0, AUG);
    // C2_cap = -(1/gamma) D22^T C2 ; D21_cap = -(1/gamma) D22^T D21 - D12^T
    gemm_s<<<g1(128*256),256,0,stream>>>(C2CAP,256, D22p,128, C2,256,
                                         128,256,128, 1,0, -1.f,2,sg, 0.f, 0);
    gemm_s<<<g1(128*256),256,0,stream>>>(D21CAP,256, D22p,128, D21,256,
                                         128,256,128, 1,0, -1.f,2,sg, 0.f, 0);
    k_addT<<<g1(128*256),256,0,stream>>>(D21CAP,256, D12,128, 128,256, -1.f);
    // vec_R = [C2_cap^T ; D21_cap^T ; B2]  (768x128)
    k_copyT<<<g1(128*256),256,0,stream>>>(VECR,            C2CAP, 128, 256);
    k_copyT<<<g1(128*256),256,0,stream>>>(VECR + 256*128,  D21CAP,128, 256);
    k_copy <<<g1(256*128),256,0,stream>>>(VECR + 512*128,  B2, 256*128);
    // vec_Q = [C2^T ; D21^T ; 0]          (768x128)
    k_copyT<<<g1(128*256),256,0,stream>>>(VECQ,            C2,  128, 256);
    k_copyT<<<g1(128*256),256,0,stream>>>(VECQ + 256*128,  D21, 128, 256);
    k_zero <<<g1(256*128),256,0,stream>>>(VECQ + 512*128,  256*128);
    // H = X^T X + EPS*I + (vecR @ Rinv) @ vecR^T + (1/gamma) vecQ @ vecQ^T
    gemm_s<<<g1(768*128),256,0,stream>>>(T1,128, VECR,128, RINV,128,
                                         768,128,128, 0,0, 1.f,0,sg, 0.f, 0);
    gemm_s<<<g1(768*768),256,0,stream>>>(Hm,768, X,768, X,768,
                                         768,768,768, 1,0, 1.f,0,sg, 0.f, 1);
    gemm_s<<<g1(768*768),256,0,stream>>>(Hm,768, T1,128, VECR,128,
                                         768,768,128, 0,1, 1.f,0,sg, 1.f, 0);
    gemm_s<<<g1(768*768),256,0,stream>>>(Hm,768, VECQ,128, VECQ,128,
                                         768,768,128, 0,1, 1.f,2,sg, 1.f, 0);
    // E = 0.5*(H11 + H33 + Y - Y^T) ; Einv
    k_E<<<g1(256*256),256,0,stream>>>(Ep, Hm, Y);
    invert_gj<<<1,1024,0,stream>>>(EINV, Ep, 256, 256, 0, AUG);

    // ---- batch phase (WMMA) ----
    // cx = xi @ C1^T + w @ D12^T,  C1 = -H21 = -H[256:512, 0:256]
    batch_gemm3<<<dim3(NXIc/64, Bb/128),256,0,stream>>>(CX, NXIc,
        xi, Hm + 256*768,       768, NXIc, -1.f,
        w,  D12,                128, Nc,    1.f,
        nullptr, nullptr, 0, 0, 0.f);
    // sequential recurrence over L (H22 = H[256:512, 256:512], ld 768)
    recur_eps<<<Bb/32, 32, 0, stream>>>(EPSb, CX, Hm + 256*768 + 256);
    // E_xi = xi @ Fm^T + eps @ B1^T + w @ B2^T   (into CX, no longer needed)
    batch_gemm3<<<dim3(NXIc/64, Bb/128),256,0,stream>>>(CX, NXIc,
        xi,   Hm + 512*768,       768, NXIc, 1.f,
        EPSb, Hm + 512*768 + 256, 768, Lc,   1.f,
        w,    B2,                 128, Nc,   1.f);
    // xi_next = E_xi @ Einv^T
    batch_gemm3<<<dim3(NXIc/64, Bb/128),256,0,stream>>>(xn_out, NXIc,
        CX, EINV, 256, NXIc, 1.f,
        nullptr, nullptr, 0, 0, 0.f,
        nullptr, nullptr, 0, 0, 0.f);
    // u = xi @ C2^T + eps @ D21^T + w @ D22^T
    batch_gemm3<<<dim3(Mc/64, Bb/128),256,0,stream>>>(u_out, Mc,
        xi,   C2,   256, NXIc, 1.f,
        EPSb, D21,  256, Lc,   1.f,
        w,    D22p, 128, Nc,   1.f);

    (void)n_in; (void)out_size; (void)ws_size;
}